// AudioFrontend_29214367547586
// MI455X (gfx1250) — compile-verified
//
#include <hip/hip_runtime.h>
#include <hip/hip_bf16.h>
#include <math.h>

// ---------------- problem constants (from reference) ----------------
#define DECIM   63
#define NFFT    512
#define HOP     160
#define WINLEN  400
#define NMELS   40
#define NPDM    60480000
#define NS      960000          // NPDM / DECIM
#define NSPAD   960512          // covers padded-frame reads: 5999*160+511 = 960351
#define H5LEN   311             // 5*(DECIM-1)+1 : support of (boxcar63)^5
#define FIRLEN  15
#define NFRAMES 5997            // 1 + (NS-NFFT)/HOP
#define MPAD    6000            // NFRAMES padded to 16
#define NBINS   257
#define NBPAD   272             // 17*16
#define NTILE_M 375             // MPAD/16
#define NTILE_N 17              // NBPAD/16
#define KMEL    260             // NBINS padded to multiple of 4
#define NMELPAD 48              // NMELS padded to 16
#define PADL    56              // (NFFT-WINLEN)/2

typedef float v2f __attribute__((ext_vector_type(2)));
typedef float v4f __attribute__((ext_vector_type(4)));
typedef float v8f __attribute__((ext_vector_type(8)));

// ---------------- h5 = coefficients of (1+z+...+z^62)^5 ----------------
// inclusion-exclusion: h5[k] = sum_j (-1)^j C(5,j) C(k-63j+4, 4)
__global__ void k_h5(int* __restrict__ h5) {
    int k = threadIdx.x;
    if (k >= H5LEN) return;
    const long long c5[6] = {1, 5, 10, 10, 5, 1};
    long long acc = 0;
    for (int j = 0; j <= 5; ++j) {
        int n = k - 63 * j;
        if (n < 0) break;
        long long m = (long long)n + 4;
        long long c = (m * (m - 1) * (m - 2) * (m - 3)) / 24;  // C(m,4)
        acc += ((j & 1) ? -c5[j] : c5[j]) * c;
    }
    h5[k] = (int)acc;
}

// ---------------- CIC decimator as direct 311-tap convolution ----------------
// xi[m] = sum_k h5[k] * (2*bit[63m-k]-1), zero for negative index.
// Bound: |xi| <= 63^5 < 2^31 -> exact in int32.
__global__ void k_cic(const int* __restrict__ bits, const int* __restrict__ h5,
                      int* __restrict__ xi) {
    __shared__ int sh5[H5LEN];
    for (int i = threadIdx.x; i < H5LEN; i += blockDim.x) sh5[i] = h5[i];
    __syncthreads();
    int m = blockIdx.x * blockDim.x + threadIdx.x;
    if (m >= NS) return;
    int base = m * DECIM;                       // < 2^31
    int kmax = (base < H5LEN - 1) ? base : (H5LEN - 1);
    const int* p = bits + base;
    __builtin_prefetch(p - kmax, 0, 1);         // global_prefetch_b8 of window tail
    __builtin_prefetch(p, 0, 1);
    int acc = 0;
    for (int k = 0; k <= kmax; ++k) {
        int b = p[-k];                          // 0 or 1
        acc += b ? sh5[k] : -sh5[k];
    }
    xi[m] = acc;
}

// ---------------- 15-tap integer compensation FIR + floor-div scale ----------------
// Also zero-fills the padded tail of y so the DFT kernel can load branch-free.
__global__ void k_fir(const int* __restrict__ xi, const long long* __restrict__ taps,
                      const int* __restrict__ scale_p, float* __restrict__ y) {
    int i = blockIdx.x * blockDim.x + threadIdx.x;
    if (i >= NSPAD) return;
    if (i >= NS || i < FIRLEN) { y[i] = 0.0f; return; }
    long long s = (long long)scale_p[0];
    long long acc = 0;
    #pragma unroll
    for (int k = 0; k < FIRLEN; ++k) acc += taps[k] * (long long)xi[i - k];
    long long q = acc / s;
    long long r = acc % s;
    if (r != 0 && ((acc < 0) != (s < 0))) q -= 1;    // floor division
    y[i] = (float)q;
}

// ---------------- windowed DFT bases, interleaved cos/sin quads ----------------
// bi[n][q] = { w*cos(2pi*n*2q/512), w*cos(2pi*n*(2q+1)/512),
//              w*sin(2pi*n*2q/512), w*sin(2pi*n*(2q+1)/512) }   (q = 0..255)
// One b128 load in the DFT loop then yields both B operands (K={2q,2q+1}).
// Sign of sin is irrelevant for |X|^2. Phase reduced exactly via (k*n) & 511.
__device__ __forceinline__ float hannwin(int k) {
    if (k < PADL || k >= PADL + WINLEN) return 0.0f;
    return 0.5f * (1.0f - cosf(6.28318530717958647692f * (float)(k - PADL) / (float)WINLEN));
}
__global__ void k_basis(float* __restrict__ bi) {
    int idx = blockIdx.x * blockDim.x + threadIdx.x;   // n*256 + q
    if (idx >= NBPAD * (NFFT / 2)) return;
    int n = idx >> 8;
    int q = idx & 255;
    int k0 = q << 1;
    float w0 = hannwin(k0), w1 = hannwin(k0 + 1);
    float a0 = (6.28318530717958647692f / (float)NFFT) * (float)((k0 * n) & (NFFT - 1));
    float a1 = (6.28318530717958647692f / (float)NFFT) * (float)(((k0 + 1) * n) & (NFFT - 1));
    v4f v;
    v.x = w0 * cosf(a0);
    v.y = w1 * cosf(a1);
    v.z = w0 * sinf(a0);
    v.w = w1 * sinf(a1);
    *(v4f*)(bi + (size_t)idx * 4) = v;                 // 16B-aligned b128 store
}

// ---------------- mel filterbank, stored transposed [mel][freq], zero padded --------
__device__ __forceinline__ double mel2hz(double m) {
    return 700.0 * (pow(10.0, m / 2595.0) - 1.0);
}
__global__ void k_fb(float* __restrict__ fbT) {
    int idx = blockIdx.x * blockDim.x + threadIdx.x;
    if (idx >= NMELPAD * KMEL) return;
    int m = idx / KMEL;
    int k = idx - m * KMEL;
    float v = 0.0f;
    if (m < NMELS && k < NBINS) {
        double fmax = 8000.0;
        double mel_max = 2595.0 * log10(1.0 + fmax / 700.0);
        double step = mel_max / (double)(NMELS + 1);
        double f0 = mel2hz(step * (double)m);
        double f1 = mel2hz(step * (double)(m + 1));
        double f2 = mel2hz(step * (double)(m + 2));
        double freq = (double)k * fmax / (double)(NFFT / 2);
        double dn = (freq - f0) / (f1 - f0);
        double up = (f2 - freq) / (f2 - f1);
        double t = dn < up ? dn : up;
        if (t < 0.0) t = 0.0;
        v = (float)t;
    }
    fbT[idx] = v;
}

// ---------------- DFT power spectrum via V_WMMA_F32_16X16X4_F32 ----------------
// One wave = one 16(frames) x 16(bins) tile; K loop over 512 time samples.
// A layout (16x4 f32): lane<16 -> M=lane, K={k0,k0+1}; lane>=16 -> M=lane-16, K={k0+2,k0+3}
// B layout (4x16 f32): mirrored. Per step: 1 b64 A-load + 1 b128 interleaved
// cos/sin B-load feeding two WMMAs (shared A).
// Branch-free: y is padded+zeroed to NSPAD; garbage in padded frame rows
// (5997..5999) only pollutes output rows that are masked at the final store.
__global__ void __launch_bounds__(256) k_dft(const float* __restrict__ y,
                                             const float* __restrict__ bi,
                                             float* __restrict__ P) {
    int wave = blockIdx.x * (blockDim.x >> 5) + (threadIdx.x >> 5);
    if (wave >= NTILE_M * NTILE_N) return;          // wave-uniform exit (EXEC stays full)
    int lane = threadIdx.x & 31;
    int mt = wave / NTILE_N, nt = wave - mt * NTILE_N;
    int f0 = mt * 16, n0 = nt * 16;
    int half = lane >> 4, col = lane & 15;
    int frame = f0 + col;
    const float* ap = y  + frame * HOP + (half << 1);
    // element offset for quad of pair q = k0/2 + half is 4q = 2*k0 + 4*half
    const float* bp = bi + (size_t)(n0 + col) * (NFFT * 2) + (half << 2);
    v8f accR = {}; v8f accI = {};
    for (int k0 = 0; k0 < NFFT; k0 += 4) {
        v2f a  = *(const v2f*)(ap + k0);
        v4f bq = *(const v4f*)(bp + (k0 << 1));     // {cos,cos,sin,sin}
        v2f vb = bq.lo;
        v2f vs = bq.hi;
        accR = __builtin_amdgcn_wmma_f32_16x16x4_f32(false, a, false, vb, (short)0, accR, false, false);
        accI = __builtin_amdgcn_wmma_f32_16x16x4_f32(false, a, false, vs, (short)0, accI, false, false);
    }
    // C/D layout: VGPR r -> row r (lanes 0-15) / row r+8 (lanes 16-31), col = lane&15
    #pragma unroll
    for (int r = 0; r < 8; ++r) {
        int fr = f0 + r + (half << 3);
        float re = accR[r], im = accI[r];
        P[fr * NBPAD + n0 + col] = re * re + im * im;
    }
}

// ---------------- mel projection + log, WMMA f32, writes transposed output -------
__global__ void __launch_bounds__(256) k_mel(const float* __restrict__ P,
                                             const float* __restrict__ fbT,
                                             float* __restrict__ out) {
    int wave = blockIdx.x * (blockDim.x >> 5) + (threadIdx.x >> 5);
    if (wave >= NTILE_M * (NMELPAD / 16)) return;
    int lane = threadIdx.x & 31;
    int mt = wave / (NMELPAD / 16), nt = wave - mt * (NMELPAD / 16);
    int f0 = mt * 16, n0 = nt * 16;
    int half = lane >> 4, col = lane & 15;
    const float* ap = P   + (f0 + col) * NBPAD + (half << 1);
    const float* bp = fbT + (n0 + col) * KMEL  + (half << 1);
    v8f acc = {};
    for (int k0 = 0; k0 < KMEL; k0 += 4) {
        v2f a = *(const v2f*)(ap + k0);
        v2f b = *(const v2f*)(bp + k0);
        acc = __builtin_amdgcn_wmma_f32_16x16x4_f32(false, a, false, b, (short)0, acc, false, false);
    }
    #pragma unroll
    for (int r = 0; r < 8; ++r) {
        int fr = f0 + r + (half << 3);
        int mc = n0 + col;
        if (fr < NFRAMES && mc < NMELS)
            out[mc * NFRAMES + fr] = logf(acc[r] + 1e-6f);
    }
}

// ---------------- host launcher ----------------
extern "C" void kernel_launch(void* const* d_in, const int* in_sizes, int n_in,
                              void* d_out, int out_size, void* d_ws, size_t ws_size,
                              hipStream_t stream) {
    (void)in_sizes; (void)n_in; (void)out_size; (void)ws_size;
    const int*       pdm   = (const int*)d_in[0];        // int32 bits 0/1
    const long long* taps  = (const long long*)d_in[1];  // int64[15]
    const int*       scale = (const int*)d_in[2];        // int scalar
    float* out = (float*)d_out;

    // workspace carve-out (~15.4 MB total)
    char* ws = (char*)d_ws;
    size_t off = 0;
    auto carve = [&](size_t bytes) -> void* {
        void* p = ws + off;
        off = (off + bytes + 255) & ~((size_t)255);
        return p;
    };
    int*   h5  = (int*)  carve((size_t)H5LEN * 4);
    int*   xi  = (int*)  carve((size_t)NS * 4);
    float* y   = (float*)carve((size_t)NSPAD * 4);
    float* bi  = (float*)carve((size_t)NBPAD * NFFT * 2 * 4);  // interleaved cos/sin
    float* fbT = (float*)carve((size_t)NMELPAD * KMEL * 4);
    float* P   = (float*)carve((size_t)MPAD * NBPAD * 4);

    k_h5   <<<1, 320, 0, stream>>>(h5);
    k_cic  <<<(NS + 255) / 256, 256, 0, stream>>>(pdm, h5, xi);
    k_fir  <<<(NSPAD + 255) / 256, 256, 0, stream>>>(xi, taps, scale, y);
    k_basis<<<(NBPAD * (NFFT / 2) + 255) / 256, 256, 0, stream>>>(bi);
    k_fb   <<<(NMELPAD * KMEL + 255) / 256, 256, 0, stream>>>(fbT);
    k_dft  <<<(NTILE_M * NTILE_N * 32 + 255) / 256, 256, 0, stream>>>(y, bi, P);
    k_mel  <<<(NTILE_M * (NMELPAD / 16) * 32 + 255) / 256, 256, 0, stream>>>(P, fbT, out);
}